// SteerableKernel_53953379172473
// MI455X (gfx1250) — compile-verified
//
#include <hip/hip_runtime.h>

typedef __attribute__((ext_vector_type(2))) float v2f;
typedef __attribute__((ext_vector_type(8))) float v8f;

#define OI    131072   // 512*256 flattened channels
#define KS    9
#define PP    81       // 9*9 pixels
#define GO    8        // group order
#define NTOT  648      // GO*PP output columns
#define KPAD  84       // 81 padded to multiple of 4 (21 k-steps)
#define NTILE 41       // ceil(656/16) column tiles (cols 648..655 are zero)

__global__ __launch_bounds__(256) void steer_rot_wmma(const float* __restrict__ w,
                                                      float* __restrict__ out) {
  __shared__ float As[16 * KPAD];     // 16 channels x 84 (5376 B)

  const int tid  = threadIdx.x;
  const int lane = tid & 31;
  const int wv   = tid >> 5;          // 8 waves per block
  const int c0   = blockIdx.x << 4;   // 16 channels per block

  // ---- stage A panel [16 x 81] -> LDS, coalesced; zero the K-pad tail ----
  const float* Ag = w + (size_t)c0 * PP;
  for (int e = tid; e < 16 * PP; e += 256)
    As[(e / PP) * KPAD + (e % PP)] = Ag[e];
  if (tid < 48)
    As[(tid / 3) * KPAD + PP + (tid % 3)] = 0.0f;
  __syncthreads();

  const int row   = lane & 15;            // A-matrix row (channel within panel)
  const int khalf = (lane >> 4) << 1;     // K sub-offset: 0 (lanes 0-15) / 2 (16-31)
  const int lm    = lane & 15;            // B/C column within tile

  for (int nt = wv; nt < NTILE; nt += 8) {
    const int  n   = (nt << 4) + lm;
    const bool ncv = (n < NTOT);
    const int  nc  = ncv ? n : 0;
    const int  g   = nc / PP;
    const int  p   = nc - g * PP;
    const int  yy  = p / KS;
    const int  xx  = p - yy * KS;

    // rotation grid, align_corners=False (matches reference math)
    float sn, cs;
    sincosf(0.78539816339744831f * (float)g, &sn, &cs);   // 2*pi*g/8
    const float lxv = ((float)xx + 0.5f) * (2.0f / 9.0f) - 1.0f;
    const float lyv = ((float)yy + 0.5f) * (2.0f / 9.0f) - 1.0f;
    const float gx  = cs * lxv - sn * lyv;
    const float gy  = sn * lxv + cs * lyv;
    const float ix  = ((gx + 1.0f) * 9.0f - 1.0f) * 0.5f;
    const float iy  = ((gy + 1.0f) * 9.0f - 1.0f) * 0.5f;
    const float fx0 = floorf(ix), fy0 = floorf(iy);
    const int   ix0 = (int)fx0,   iy0 = (int)fy0;
    const float wx1 = ix - fx0, wx0 = 1.0f - wx1;
    const float wy1 = iy - fy0, wy0 = 1.0f - wy1;

    int   kk0, kk1, kk2, kk3;
    float tw0, tw1, tw2, tw3;
    auto tap = [&](int yi, int xi, float wgt, int& kt, float& wt) {
      bool v = ncv && ((unsigned)xi < 9u) && ((unsigned)yi < 9u);  // zeros padding
      kt = v ? (yi * KS + xi) : -1;
      wt = v ? wgt : 0.0f;
    };
    tap(iy0,     ix0,     wy0 * wx0, kk0, tw0);
    tap(iy0,     ix0 + 1, wy0 * wx1, kk1, tw1);
    tap(iy0 + 1, ix0,     wy1 * wx0, kk2, tw2);
    tap(iy0 + 1, ix0 + 1, wy1 * wx1, kk3, tw3);

    // which K=4 blocks does this lane's column touch?
    unsigned kkbits = 0;
    kkbits |= (kk0 >= 0) ? (1u << (kk0 >> 2)) : 0u;
    kkbits |= (kk1 >= 0) ? (1u << (kk1 >> 2)) : 0u;
    kkbits |= (kk2 >= 0) ? (1u << (kk2 >> 2)) : 0u;
    kkbits |= (kk3 >= 0) ? (1u << (kk3 >> 2)) : 0u;

    v8f acc = {0.f, 0.f, 0.f, 0.f, 0.f, 0.f, 0.f, 0.f};

    for (int kk = 0; kk < 21; ++kk) {
      // wave-uniform skip of all-zero K blocks (EXEC stays all-ones for WMMA)
      if (__builtin_amdgcn_ballot_w32((kkbits >> kk) & 1u) == 0u) continue;

      const int ka = (kk << 2) + khalf;
      const int kb = ka + 1;

      // A fragment from LDS (8B-aligned, conflict-free stride 84)
      v2f a = *(const v2f*)&As[row * KPAD + ka];

      // B fragment rebuilt from the (<=4) bilinear taps of this column
      v2f b;
      b.x = (ka == kk0 ? tw0 : 0.0f) + (ka == kk1 ? tw1 : 0.0f)
          + (ka == kk2 ? tw2 : 0.0f) + (ka == kk3 ? tw3 : 0.0f);
      b.y = (kb == kk0 ? tw0 : 0.0f) + (kb == kk1 ? tw1 : 0.0f)
          + (kb == kk2 ? tw2 : 0.0f) + (kb == kk3 ? tw3 : 0.0f);

      // full-precision fp32 matrix op: D = A(16x4) * B(4x16) + C
      acc = __builtin_amdgcn_wmma_f32_16x16x4_f32(false, a, false, b,
                                                  (short)0, acc, false, false);
    }

    // C/D layout: VGPR j -> row j (lanes 0-15) / row j+8 (lanes 16-31), col = lane%16
    if (ncv) {
      const size_t base = (size_t)g * ((size_t)OI * PP) + (size_t)p;
      const int    crow = c0 + ((lane >> 4) << 3);
#pragma unroll
      for (int j = 0; j < 8; ++j)
        out[base + (size_t)(crow + j) * PP] = acc[j];
    }
  }
}

extern "C" void kernel_launch(void* const* d_in, const int* in_sizes, int n_in,
                              void* d_out, int out_size, void* d_ws, size_t ws_size,
                              hipStream_t stream) {
  (void)in_sizes; (void)n_in; (void)d_ws; (void)ws_size; (void)out_size;
  const float* wptr = (const float*)d_in[0];   // base_weight [512,256,9,9] fp32
  float* outp = (float*)d_out;                 // [8*512,256,9,9] fp32
  steer_rot_wmma<<<OI / 16, 256, 0, stream>>>(wptr, outp);
}